// KoLeoLoss_71683004170507
// MI455X (gfx1250) — compile-verified
//
#include <hip/hip_runtime.h>
#include <math.h>

#define NROWS 16384
#define NDIM  256
#define LDS_STRIDE 260   // 256 + 4 pad: distinct LDS banks for the 16 row-lanes, 16B aligned
#define KEPS 1e-8f

typedef __attribute__((ext_vector_type(2))) float v2f;
typedef __attribute__((ext_vector_type(8))) float v8f;

// ---------------------------------------------------------------------------
// Kernel 1: fused Gram-matrix + per-row argmax (diagonal masked).
// One wave owns a 16-row tile (A fragments hoisted into registers by the
// compiler after one LDS staging pass) and loops over column tiles TWO at a
// time: two independent v_wmma_f32_16x16x4_f32 accumulation chains reuse the
// same A registers, doubling MLP and breaking the single-C dependency chain.
// ---------------------------------------------------------------------------
__global__ __launch_bounds__(64) void koleo_argmax_kernel(
    const float* __restrict__ x, int* __restrict__ nn_idx)
{
    __shared__ float lds[2 * 16 * LDS_STRIDE];

    const int lane = threadIdx.x & 31;
    const int wave = threadIdx.x >> 5;
    const int row_base = (blockIdx.x * 2 + wave) * 16;
    float* atile = &lds[wave * 16 * LDS_STRIDE];

    // Stage this wave's 16x256 A tile into padded LDS rows (float4 stores).
    for (int t = lane; t < 16 * (NDIM / 4); t += 32) {
        const int r  = t >> 6;        // 64 float4 per row
        const int c4 = t & 63;
        const float4 v = ((const float4*)(x + (size_t)(row_base + r) * NDIM))[c4];
        *(float4*)&atile[r * LDS_STRIDE + c4 * 4] = v;
    }
    __syncthreads();

    // WMMA 32-bit fragment layout: lanes 0-15 hold K={0,1}, lanes 16-31 K={2,3}.
    const int half = lane >> 4;       // 0 or 1
    const int l16  = lane & 15;
    const int koff = half * 2;
    const int rowg = row_base + half * 8;   // rows held by this lane's C VGPRs
    const float* arow = &atile[l16 * LDS_STRIDE + koff];

    float bestv[8];
    int   besti[8];
#pragma unroll
    for (int r = 0; r < 8; ++r) { bestv[r] = -2.0f; besti[r] = 0; }

    for (int ct = 0; ct < NROWS / 16; ct += 2) {
        const int col0 = ct * 16 + l16;               // B fragment: column = lane%16
        const int col1 = col0 + 16;
        const float* brow0 = x + (size_t)col0 * NDIM + koff;
        const float* brow1 = x + (size_t)col1 * NDIM + koff;

        v8f c0 = {0.f, 0.f, 0.f, 0.f, 0.f, 0.f, 0.f, 0.f};
        v8f c1 = {0.f, 0.f, 0.f, 0.f, 0.f, 0.f, 0.f, 0.f};
#pragma unroll
        for (int kc = 0; kc < NDIM / 4; ++kc) {
            const v2f a  = *(const v2f*)(arow + kc * 4);   // reg-resident after hoist
            const v2f b0 = *(const v2f*)(brow0 + kc * 4);  // global_load_b64
            const v2f b1 = *(const v2f*)(brow1 + kc * 4);  // global_load_b64
            c0 = __builtin_amdgcn_wmma_f32_16x16x4_f32(
                     false, a, false, b0, (short)0, c0, false, false);
            c1 = __builtin_amdgcn_wmma_f32_16x16x4_f32(
                     false, a, false, b1, (short)0, c1, false, false);
        }

        // Running argmax. C layout: VGPR r -> row (rowg + r), column = col.
#pragma unroll
        for (int r = 0; r < 8; ++r) {
            const float v0 = c0[r];
            if (col0 != (rowg + r) && v0 > bestv[r]) { bestv[r] = v0; besti[r] = col0; }
            const float v1 = c1[r];
            if (col1 != (rowg + r) && v1 > bestv[r]) { bestv[r] = v1; besti[r] = col1; }
        }
    }

    // Reduce across the 16 lanes that share each row (keep lowest index on ties).
#pragma unroll
    for (int r = 0; r < 8; ++r) {
        float v = bestv[r];
        int   idx = besti[r];
        for (int m = 8; m >= 1; m >>= 1) {
            const float ov = __shfl_xor(v, m, 16);
            const int   oi = __shfl_xor(idx, m, 16);
            if (ov > v || (ov == v && oi < idx)) { v = ov; idx = oi; }
        }
        if (l16 == 0) nn_idx[rowg + r] = idx;
    }
}

// ---------------------------------------------------------------------------
// Kernel 2: per-row ||x_i - x_nn + eps||_2  ->  -log(dist + eps)
// One wave per row; lane covers 8 of the 256 dims (two float4 loads).
// ---------------------------------------------------------------------------
__global__ __launch_bounds__(256) void koleo_dist_kernel(
    const float* __restrict__ x, const int* __restrict__ nn_idx,
    float* __restrict__ rowloss)
{
    const int lane = threadIdx.x & 31;
    const int i    = blockIdx.x * 8 + (threadIdx.x >> 5);
    const int j    = nn_idx[i];

    const float4* xi = (const float4*)(x + (size_t)i * NDIM);
    const float4* xj = (const float4*)(x + (size_t)j * NDIM);

    float acc = 0.f;
#pragma unroll
    for (int t = 0; t < 2; ++t) {
        const float4 a = xi[lane * 2 + t];
        const float4 b = xj[lane * 2 + t];
        const float d0 = a.x - b.x + KEPS, d1 = a.y - b.y + KEPS;
        const float d2 = a.z - b.z + KEPS, d3 = a.w - b.w + KEPS;
        acc += d0 * d0 + d1 * d1 + d2 * d2 + d3 * d3;
    }
    for (int m = 16; m >= 1; m >>= 1) acc += __shfl_xor(acc, m, 32);
    if (lane == 0) rowloss[i] = -logf(sqrtf(acc) + KEPS);
}

// ---------------------------------------------------------------------------
// Kernel 3: deterministic single-block mean reduction of the 16384 row losses.
// ---------------------------------------------------------------------------
__global__ __launch_bounds__(256) void koleo_reduce_kernel(
    const float* __restrict__ rowloss, float* __restrict__ out)
{
    __shared__ float s[256];
    float acc = 0.f;
    for (int t = threadIdx.x; t < NROWS; t += 256) acc += rowloss[t];
    s[threadIdx.x] = acc;
    __syncthreads();
    for (int w = 128; w >= 1; w >>= 1) {
        if ((int)threadIdx.x < w) s[threadIdx.x] += s[threadIdx.x + w];
        __syncthreads();
    }
    if (threadIdx.x == 0) out[0] = s[0] / (float)NROWS;
}

extern "C" void kernel_launch(void* const* d_in, const int* in_sizes, int n_in,
                              void* d_out, int out_size, void* d_ws, size_t ws_size,
                              hipStream_t stream)
{
    const float* x = (const float*)d_in[0];

    int*   nn      = (int*)d_ws;                                   // 64 KB
    float* rowloss = (float*)((char*)d_ws + NROWS * sizeof(int));  // 64 KB

    // 512 blocks x 64 threads (2 waves each) -> 1024 row-tiles of 16.
    koleo_argmax_kernel<<<NROWS / 32, 64, 0, stream>>>(x, nn);
    // One wave per row: 2048 blocks x 8 waves.
    koleo_dist_kernel<<<NROWS / 8, 256, 0, stream>>>(x, nn, rowloss);
    // Deterministic final mean.
    koleo_reduce_kernel<<<1, 256, 0, stream>>>(rowloss, (float*)d_out);
}